// NodeUpdate_26740466385760
// MI455X (gfx1250) — compile-verified
//
#include <hip/hip_runtime.h>

#define NODE_DIM 128
#define EDGE_DIM 48
#define HIDDEN   64
#define IN_DIM   176          // 128 + 48
#define XS       180          // padded LDS row stride (floats): conflict-free for col reads
#define KG1      44           // IN_DIM / 4  (k-groups for GEMM1)
#define KG2      16           // HIDDEN / 4  (k-groups for GEMM2/3)
#define LN_EPS   1e-5f

typedef float v2f __attribute__((ext_vector_type(2)));
typedef float v8f __attribute__((ext_vector_type(8)));

// ---------------------------------------------------------------------------
// Kernel 1: scatter-add edge features + degree count.
// Thread t handles one float4 chunk (12 chunks per 48-float edge row).
// Reads are perfectly coalesced; adds are global_atomic_add_f32 (no-return)
// resolved at L2 (19 MB agg buffer << 192 MB L2).
// ---------------------------------------------------------------------------
__global__ __launch_bounds__(256)
void scatter_kernel(const float* __restrict__ ef, const int* __restrict__ recv,
                    float* __restrict__ agg, float* __restrict__ deg, int n_edges) {
    int gid = blockIdx.x * blockDim.x + threadIdx.x;
    int total = n_edges * 12;
    if (gid >= total) return;
    int e = gid / 12, c = gid % 12;
    int r = recv[e];
    const float4 v = *reinterpret_cast<const float4*>(ef + (size_t)e * EDGE_DIM + c * 4);
    float* dst = agg + (size_t)r * EDGE_DIM + c * 4;
    unsafeAtomicAdd(dst + 0, v.x);
    unsafeAtomicAdd(dst + 1, v.y);
    unsafeAtomicAdd(dst + 2, v.z);
    unsafeAtomicAdd(dst + 3, v.w);
    if (c == 0) unsafeAtomicAdd(deg + r, 1.0f);
}

// ---------------------------------------------------------------------------
// Kernel 2: swizzle W1/W2/W3 into per-lane B-fragment order so GEMM B loads
// are a single coalesced global_load_b64 per wave per (k-group, n-tile).
// Entry s -> (g, t, lane): O[2s..2s+1] = { W[k0][col], W[k0+1][col] }
//   col = 16t + (lane&15), k0 = 4g + 2*(lane>=16)
// ---------------------------------------------------------------------------
__device__ inline void swz_one(const float* __restrict__ W, float* __restrict__ O, int s) {
    int g   = s >> 7;          // s / 128
    int rem = s & 127;
    int t   = rem >> 5;
    int L   = rem & 31;
    int col = t * 16 + (L & 15);
    int k0  = 4 * g + ((L >> 4) << 1);
    O[2 * s]     = W[(size_t)k0 * HIDDEN + col];
    O[2 * s + 1] = W[(size_t)(k0 + 1) * HIDDEN + col];
}

__global__ __launch_bounds__(256)
void prep_weights(const float* __restrict__ W1, const float* __restrict__ W2,
                  const float* __restrict__ W3,
                  float* __restrict__ w1s, float* __restrict__ w2s, float* __restrict__ w3s) {
    int s = blockIdx.x * blockDim.x + threadIdx.x;
    const int n1 = KG1 * 128;          // 5632
    const int n2 = KG2 * 128;          // 2048
    if (s < n1)                swz_one(W1, w1s, s);
    else if (s < n1 + n2)      swz_one(W2, w2s, s - n1);
    else if (s < n1 + 2 * n2)  swz_one(W3, w3s, s - n1 - n2);
}

// ---------------------------------------------------------------------------
// Kernel 3: per-16-node-tile MLP + LayerNorm, one wave per tile,
// V_WMMA_F32_16X16X4_F32 for all three GEMMs.
// ---------------------------------------------------------------------------
__global__ __launch_bounds__(128)
void node_mlp_kernel(const float* __restrict__ nodes, const float* __restrict__ agg,
                     const float* __restrict__ deg,
                     const float* __restrict__ w1s, const float* __restrict__ w2s,
                     const float* __restrict__ w3s,
                     const float* __restrict__ b1, const float* __restrict__ b2,
                     const float* __restrict__ b3,
                     const float* __restrict__ gamma, const float* __restrict__ beta,
                     float* __restrict__ out, int n_nodes) {
    __shared__ float Xs[4][16 * XS];

    const int lane  = threadIdx.x & 31;
    const int wave  = threadIdx.x >> 5;
    float* Xw = Xs[wave];

    const int tile   = blockIdx.x * 4 + wave;
    const int base   = tile * 16;
    const bool active = (base < n_nodes);
    const int base_c = active ? base : 0;

    const int arow  = lane & 15;          // M (A-matrix row / N column within tile)
    const int ahalf = (lane >> 4) << 1;   // K offset: 0 for lanes 0-15, 2 for 16-31
    const int mbase = (lane >> 4) << 3;   // C/D: M offset 0 or 8

    // ---- stage X = [node_features | agg/deg] into LDS (wave-private tile) ----
    for (int i = lane; i < 16 * 44; i += 32) {
        int row = i / 44, c4 = i % 44;
        int g = base_c + row;
        if (g >= n_nodes) g = n_nodes - 1;
        float4 v;
        if (c4 < 32) {
            v = *reinterpret_cast<const float4*>(nodes + (size_t)g * NODE_DIM + c4 * 4);
        } else {
            int j = c4 - 32;
            v = *reinterpret_cast<const float4*>(agg + (size_t)g * EDGE_DIM + j * 4);
            float d = deg[g];
            d = d < 1.0f ? 1.0f : d;
            float inv = 1.0f / d;
            v.x *= inv; v.y *= inv; v.z *= inv; v.w *= inv;
        }
        *reinterpret_cast<float4*>(Xw + row * XS + c4 * 4) = v;
    }
    __builtin_amdgcn_wave_barrier();
    asm volatile("s_wait_dscnt 0" ::: "memory");   // wave-private LDS RAW fence

    v8f acc[4];
    v8f zero = {};

    // -------------------- GEMM1: [16x176] @ [176x64] --------------------
    acc[0] = zero; acc[1] = zero; acc[2] = zero; acc[3] = zero;
    for (int g = 0; g < KG1; ++g) {
        v2f a = *reinterpret_cast<const v2f*>(Xw + arow * XS + 4 * g + ahalf);
        const float* wp = w1s + ((size_t)(g * 4) * 32 + lane) * 2;
#pragma unroll
        for (int t = 0; t < 4; ++t) {
            v2f b = *reinterpret_cast<const v2f*>(wp + t * 64);
            acc[t] = __builtin_amdgcn_wmma_f32_16x16x4_f32(
                false, a, false, b, (short)0, acc[t], false, false);
        }
    }
    // bias + ReLU + writeback H1 -> LDS rows (stride XS)
#pragma unroll
    for (int t = 0; t < 4; ++t) {
        float bb = b1[t * 16 + arow];
#pragma unroll
        for (int r = 0; r < 8; ++r) {
            float h = acc[t][r] + bb;
            h = h > 0.0f ? h : 0.0f;
            Xw[(mbase + r) * XS + t * 16 + arow] = h;
        }
    }
    __builtin_amdgcn_wave_barrier();
    asm volatile("s_wait_dscnt 0" ::: "memory");

    // -------------------- GEMM2: [16x64] @ [64x64] --------------------
    acc[0] = zero; acc[1] = zero; acc[2] = zero; acc[3] = zero;
    for (int g = 0; g < KG2; ++g) {
        v2f a = *reinterpret_cast<const v2f*>(Xw + arow * XS + 4 * g + ahalf);
        const float* wp = w2s + ((size_t)(g * 4) * 32 + lane) * 2;
#pragma unroll
        for (int t = 0; t < 4; ++t) {
            v2f b = *reinterpret_cast<const v2f*>(wp + t * 64);
            acc[t] = __builtin_amdgcn_wmma_f32_16x16x4_f32(
                false, a, false, b, (short)0, acc[t], false, false);
        }
    }
#pragma unroll
    for (int t = 0; t < 4; ++t) {
        float bb = b2[t * 16 + arow];
#pragma unroll
        for (int r = 0; r < 8; ++r) {
            float h = acc[t][r] + bb;
            h = h > 0.0f ? h : 0.0f;
            Xw[(mbase + r) * XS + t * 16 + arow] = h;
        }
    }
    __builtin_amdgcn_wave_barrier();
    asm volatile("s_wait_dscnt 0" ::: "memory");

    // -------------------- GEMM3: [16x64] @ [64x64] (no ReLU) --------------------
    acc[0] = zero; acc[1] = zero; acc[2] = zero; acc[3] = zero;
    for (int g = 0; g < KG2; ++g) {
        v2f a = *reinterpret_cast<const v2f*>(Xw + arow * XS + 4 * g + ahalf);
        const float* wp = w3s + ((size_t)(g * 4) * 32 + lane) * 2;
#pragma unroll
        for (int t = 0; t < 4; ++t) {
            v2f b = *reinterpret_cast<const v2f*>(wp + t * 64);
            acc[t] = __builtin_amdgcn_wmma_f32_16x16x4_f32(
                false, a, false, b, (short)0, acc[t], false, false);
        }
    }
#pragma unroll
    for (int t = 0; t < 4; ++t) {
        float bb = b3[t * 16 + arow];
        acc[t] = acc[t] + bb;   // scalar splat add over v8f
    }

    // -------------------- LayerNorm over 64 (per row M) --------------------
    float gam[4], bet[4];
#pragma unroll
    for (int t = 0; t < 4; ++t) {
        gam[t] = gamma[t * 16 + arow];
        bet[t] = beta[t * 16 + arow];
    }
#pragma unroll
    for (int r = 0; r < 8; ++r) {
        float h0 = acc[0][r], h1 = acc[1][r], h2 = acc[2][r], h3 = acc[3][r];
        float s = h0 + h1 + h2 + h3;
        float q = h0 * h0 + h1 * h1 + h2 * h2 + h3 * h3;
        // reduce over the 16 lanes that hold this row
        s += __shfl_xor(s, 1, 16);  q += __shfl_xor(q, 1, 16);
        s += __shfl_xor(s, 2, 16);  q += __shfl_xor(q, 2, 16);
        s += __shfl_xor(s, 4, 16);  q += __shfl_xor(q, 4, 16);
        s += __shfl_xor(s, 8, 16);  q += __shfl_xor(q, 8, 16);
        float mu  = s * (1.0f / 64.0f);
        float var = q * (1.0f / 64.0f) - mu * mu;
        float rs  = rsqrtf(var + LN_EPS);
        int row = base + mbase + r;
        if (active && row < n_nodes) {
            float* op = out + (size_t)row * HIDDEN + arow;
            op[0]  = (h0 - mu) * rs * gam[0] + bet[0];
            op[16] = (h1 - mu) * rs * gam[1] + bet[1];
            op[32] = (h2 - mu) * rs * gam[2] + bet[2];
            op[48] = (h3 - mu) * rs * gam[3] + bet[3];
        }
    }
}

// ---------------------------------------------------------------------------
extern "C" void kernel_launch(void* const* d_in, const int* in_sizes, int n_in,
                              void* d_out, int out_size, void* d_ws, size_t ws_size,
                              hipStream_t stream) {
    const float* nodes = (const float*)d_in[0];
    const float* edges = (const float*)d_in[1];
    const int*   recv  = (const int*)d_in[2];
    const float* W1    = (const float*)d_in[4];
    const float* b1    = (const float*)d_in[5];
    const float* W2    = (const float*)d_in[6];
    const float* b2    = (const float*)d_in[7];
    const float* W3    = (const float*)d_in[8];
    const float* b3    = (const float*)d_in[9];
    const float* gamma = (const float*)d_in[10];
    const float* beta  = (const float*)d_in[11];

    const int n_nodes = in_sizes[0] / NODE_DIM;
    const int n_edges = in_sizes[2];

    float* ws  = (float*)d_ws;
    float* agg = ws;                                 // n_nodes * 48
    float* deg = agg + (size_t)n_nodes * EDGE_DIM;   // n_nodes
    float* w1s = deg + n_nodes;                      // 44*128*2 = 11264
    float* w2s = w1s + KG1 * 128 * 2;                // 16*128*2 = 4096
    float* w3s = w2s + KG2 * 128 * 2;                // 4096

    // zero agg + deg (adjacent): n_nodes * 49 floats
    hipMemsetAsync(agg, 0, (size_t)n_nodes * 49 * sizeof(float), stream);

    // swizzle weights into B-fragment order
    const int n_swz = (KG1 + 2 * KG2) * 128;         // 9728
    prep_weights<<<(n_swz + 255) / 256, 256, 0, stream>>>(W1, W2, W3, w1s, w2s, w3s);

    // scatter edges
    const int total = n_edges * 12;
    scatter_kernel<<<(total + 255) / 256, 256, 0, stream>>>(edges, recv, agg, deg, n_edges);

    // node MLP + LN
    const int tiles = (n_nodes + 15) / 16;
    node_mlp_kernel<<<(tiles + 3) / 4, 128, 0, stream>>>(
        nodes, agg, deg, w1s, w2s, w3s, b1, b2, b3, gamma, beta,
        (float*)d_out, n_nodes);
}